// SSIM_45655502357282
// MI455X (gfx1250) — compile-verified
//
#include <hip/hip_runtime.h>

typedef __attribute__((ext_vector_type(2))) float v2f;
typedef __attribute__((ext_vector_type(8))) float v8f;

#define TILE   64
#define HALO   5
#define REG    74            // TILE + 2*HALO
#define IMG    512
#define PLANE  (512*512)
#define HSZ    (REG*TILE)    // 74*64 = 4736 floats per blurred field

#define W_OFF  0
#define A_OFF  16
#define B_OFF  (A_OFF + REG*REG)
#define H_OFF  (B_OFF + REG*REG)
#define SMEM_FLOATS (H_OFF + 5*HSZ)   // 16 + 2*5476 + 5*4736 = 34648 floats = 138592 B

#define NBLOCKS 2048          // 32 images * 8*8 tiles
#define NPIX    8388608.0f    // 32*512*512

__global__ void SSIM_tile_kernel(const float* __restrict__ x,
                                 const float* __restrict__ y,
                                 float* __restrict__ partial)
{
    extern __shared__ float smem[];
    const int tid = threadIdx.x;
    const int blk = blockIdx.x;
    const int n   = blk >> 6;          // image
    const int t   = blk & 63;          // tile within image
    const int r0  = (t >> 3) * TILE;
    const int c0  = (t & 7) * TILE;

    // 11-tap Gaussian, sigma=1.5, normalized (matches reference kernel)
    const float Wg[11] = {
        0.00102848f, 0.00759893f, 0.03600081f, 0.10936083f, 0.21300541f,
        0.26601172f,
        0.21300541f, 0.10936083f, 0.03600081f, 0.00759893f, 0.00102848f
    };
    if (tid < 11) smem[W_OFF + tid] = Wg[tid];

    // ---- Stage A: load halo region, fuse RGB->Y, zero-pad outside image ----
    const size_t imgbase = (size_t)n * 3 * PLANE;
    for (int i = tid; i < REG * REG; i += 256) {
        const int row = i / REG;
        const int col = i - row * REG;
        const int gr  = r0 + row - HALO;
        const int gc  = c0 + col - HALO;
        float av = 0.0f, bv = 0.0f;
        if ((unsigned)gr < IMG && (unsigned)gc < IMG) {
            const size_t p = imgbase + (size_t)gr * IMG + (size_t)gc;
            const float xr = x[p], xg = x[p + PLANE], xb = x[p + 2 * PLANE];
            const float yr = y[p], yg = y[p + PLANE], yb = y[p + 2 * PLANE];
            av = 0.299f * xr + 0.587f * xg + 0.114f * xb;
            bv = 0.299f * yr + 0.587f * yg + 0.114f * yb;
        }
        smem[A_OFF + i] = av;
        smem[B_OFF + i] = bv;
    }
    __syncthreads();

    // ---- Stage B: horizontal blur of a, b, a^2, b^2, ab -> 5 fields in LDS ----
    for (int i = tid; i < REG * TILE; i += 256) {
        const int row = i >> 6;           // / TILE
        const int col = i & (TILE - 1);
        const float* ar = smem + A_OFF + row * REG + col;
        const float* br = smem + B_OFF + row * REG + col;
        float ha = 0.f, hb = 0.f, haa = 0.f, hbb = 0.f, hab = 0.f;
#pragma unroll
        for (int k = 0; k < 11; ++k) {
            const float av = ar[k], bv = br[k], wk = Wg[k];
            ha  += wk * av;
            hb  += wk * bv;
            haa += wk * av * av;
            hbb += wk * bv * bv;
            hab += wk * av * bv;
        }
        smem[H_OFF + 0 * HSZ + i] = ha;
        smem[H_OFF + 1 * HSZ + i] = hb;
        smem[H_OFF + 2 * HSZ + i] = haa;
        smem[H_OFF + 3 * HSZ + i] = hbb;
        smem[H_OFF + 4 * HSZ + i] = hab;
    }
    __syncthreads();

    // ---- Stage C: vertical blur as WMMA matmul: Out(16x16) = W(16x26) x H(26x16) ----
    // A-matrix (16x4 f32 chunks): lanes 0-15 hold K0/K1, lanes 16-31 hold K2/K3.
    const int lane = tid & 31;
    const int half = lane >> 4;        // 0 or 1
    const int m    = lane & 15;        // A row / B column within half
    v2f Aj[7];
#pragma unroll
    for (int j = 0; j < 7; ++j) {
#pragma unroll
        for (int tt = 0; tt < 2; ++tt) {
            const int K = 4 * j + 2 * half + tt;
            const int d = K - m;                       // band offset
            Aj[j][tt] = (d >= 0 && d <= 10) ? smem[W_OFF + d] : 0.0f;
        }
    }

    float lsum = 0.0f;
    const int wv = tid >> 5;                 // wave id, 0..7
#pragma unroll
    for (int s = 0; s < 2; ++s) {
        const int st  = wv * 2 + s;          // subtile 0..15
        const int sr  = (st >> 2) * 16;      // output row of subtile in tile
        const int sc  = (st & 3) * 16;       // output col of subtile in tile
        const int col = sc + m;

        v8f acc[5];
#pragma unroll
        for (int f = 0; f < 5; ++f) {
            const float* hf = smem + H_OFF + f * HSZ;
            v8f c = {};
#pragma unroll
            for (int j = 0; j < 7; ++j) {
                int k0 = 4 * j + 2 * half;
                int k1 = k0 + 1;
                k0 = k0 > 25 ? 25 : k0;      // K=26,27 have zero weight in A; clamp addr
                k1 = k1 > 25 ? 25 : k1;
                v2f b;
                b[0] = hf[(sr + k0) * TILE + col];
                b[1] = hf[(sr + k1) * TILE + col];
                c = __builtin_amdgcn_wmma_f32_16x16x4_f32(
                        false, Aj[j], false, b, (short)0, c, false, false);
            }
            acc[f] = c;
        }

        // ---- SSIM map + clipped loss, elementwise on accumulator registers ----
#pragma unroll
        for (int e = 0; e < 8; ++e) {
            const float mu1 = acc[0][e];
            const float mu2 = acc[1][e];
            const float s1  = acc[2][e] - mu1 * mu1;
            const float s2  = acc[3][e] - mu2 * mu2;
            const float s12 = acc[4][e] - mu1 * mu2;
            const float num = (2.0f * mu1 * mu2 + 1.0e-4f) * (2.0f * s12 + 9.0e-4f);
            const float den = (mu1 * mu1 + mu2 * mu2 + 1.0e-4f) * (s1 + s2 + 9.0e-4f);
            float v = 1.0f - num / den;
            v = fminf(fmaxf(v, 0.0f), 1.0f);
            lsum += v;
        }
    }

    // ---- deterministic block reduction ----
    __syncthreads();
    float* red = smem;                 // reuse LDS (weights/a-region no longer needed)
    red[tid] = lsum;
    __syncthreads();
#pragma unroll
    for (int off = 128; off > 0; off >>= 1) {
        if (tid < off) red[tid] += red[tid + off];
        __syncthreads();
    }
    if (tid == 0) partial[blk] = red[0];
}

__global__ void SSIM_final_kernel(const float* __restrict__ partial,
                                  float* __restrict__ out)
{
    __shared__ float red[256];
    const int tid = threadIdx.x;
    float s = 0.0f;
    for (int i = tid; i < NBLOCKS; i += 256) s += partial[i];
    red[tid] = s;
    __syncthreads();
#pragma unroll
    for (int off = 128; off > 0; off >>= 1) {
        if (tid < off) red[tid] += red[tid + off];
        __syncthreads();
    }
    // result = 1 - 2*loss, loss = mean(clip(1-ssim)*0.5)  =>  1 - sum/N
    if (tid == 0) out[0] = 1.0f - red[0] * (1.0f / NPIX);
}

extern "C" void kernel_launch(void* const* d_in, const int* in_sizes, int n_in,
                              void* d_out, int out_size, void* d_ws, size_t ws_size,
                              hipStream_t stream)
{
    const float* x = (const float*)d_in[0];
    const float* y = (const float*)d_in[1];
    float* partial = (float*)d_ws;     // NBLOCKS floats of scratch

    const size_t shmem = SMEM_FLOATS * sizeof(float);
    SSIM_tile_kernel<<<dim3(NBLOCKS), dim3(256), shmem, stream>>>(x, y, partial);
    SSIM_final_kernel<<<dim3(1), dim3(256), 0, stream>>>(partial, (float*)d_out);
}